// Propagate_43293270343695
// MI455X (gfx1250) — compile-verified
//
#include <hip/hip_runtime.h>
#include <hip/hip_bf16.h>

typedef __attribute__((ext_vector_type(2))) float v2f;
typedef __attribute__((ext_vector_type(8))) float v8f;

#define D 128
#define ROWS_PER_BLOCK 256   // 8 waves x 2 stripes x 16 rows

// ---------------------------------------------------------------------------
// Zero a float buffer (grid-stride).
// ---------------------------------------------------------------------------
__global__ __launch_bounds__(256) void zero_f(float* __restrict__ p, long n) {
    long i = (long)blockIdx.x * blockDim.x + threadIdx.x;
    long stride = (long)gridDim.x * blockDim.x;
    for (; i < n; i += stride) p[i] = 0.0f;
}

// ---------------------------------------------------------------------------
// Prep the four 128x128 matrices:
//   M_uu = H_ui @ H_ui^T      M_ii = H_iu @ H_iu^T
//   A1   = H_ui + H_iu^T      A2   = H_iu + H_ui^T
// ---------------------------------------------------------------------------
__global__ __launch_bounds__(256) void prep_mats(const float* __restrict__ Hui,
                                                 const float* __restrict__ Hiu,
                                                 float* __restrict__ Muu,
                                                 float* __restrict__ Mii,
                                                 float* __restrict__ A1,
                                                 float* __restrict__ A2) {
    int idx = blockIdx.x * blockDim.x + threadIdx.x;   // 0..16383
    int i = idx >> 7;
    int j = idx & 127;
    const float* hui_i = Hui + i * D;
    const float* hui_j = Hui + j * D;
    const float* hiu_i = Hiu + i * D;
    const float* hiu_j = Hiu + j * D;
    float su = 0.0f, si = 0.0f;
#pragma unroll 8
    for (int k = 0; k < D; ++k) {
        su = fmaf(hui_i[k], hui_j[k], su);
        si = fmaf(hiu_i[k], hiu_j[k], si);
    }
    Muu[idx] = su;
    Mii[idx] = si;
    A1[idx] = Hui[i * D + j] + Hiu[j * D + i];
    A2[idx] = Hiu[i * D + j] + Hui[j * D + i];
}

// ---------------------------------------------------------------------------
// Edge scatter: one wave32 per edge.
//   etype1: T_item[dst1] += w1 * Y_user[src1];  dp1[src1] += 1
//   etype2: T_user[dst2] += w2 * Y_item[src2];  dp2[src2] += 1
// ---------------------------------------------------------------------------
__global__ __launch_bounds__(256) void edge_scatter(const float* __restrict__ Yu,
                                                    const float* __restrict__ Yi,
                                                    const int* __restrict__ src1,
                                                    const int* __restrict__ dst1,
                                                    const int* __restrict__ src2,
                                                    const int* __restrict__ dst2,
                                                    const float* __restrict__ w1,
                                                    const float* __restrict__ w2,
                                                    float* __restrict__ dp1,
                                                    float* __restrict__ dp2,
                                                    float* __restrict__ Titem,
                                                    float* __restrict__ Tuser,
                                                    int E) {
    long gid = (long)blockIdx.x * blockDim.x + threadIdx.x;
    int wave = (int)(gid >> 5);
    int lane = (int)(gid & 31);
    if (wave < E) {
        int s = src1[wave];
        int d = dst1[wave];
        float wt = w1[wave];
        const float* srow = Yu + (long)s * D + lane * 4;
        __builtin_prefetch(srow, 0, 1);
        float4 v = *(const float4*)srow;
        float* drow = Titem + (long)d * D + lane * 4;
        unsafeAtomicAdd(drow + 0, wt * v.x);
        unsafeAtomicAdd(drow + 1, wt * v.y);
        unsafeAtomicAdd(drow + 2, wt * v.z);
        unsafeAtomicAdd(drow + 3, wt * v.w);
        if (lane == 0) unsafeAtomicAdd(dp1 + s, 1.0f);
    } else if (wave < 2 * E) {
        int e = wave - E;
        int s = src2[e];
        int d = dst2[e];
        float wt = w2[e];
        const float* srow = Yi + (long)s * D + lane * 4;
        __builtin_prefetch(srow, 0, 1);
        float4 v = *(const float4*)srow;
        float* drow = Tuser + (long)d * D + lane * 4;
        unsafeAtomicAdd(drow + 0, wt * v.x);
        unsafeAtomicAdd(drow + 1, wt * v.y);
        unsafeAtomicAdd(drow + 2, wt * v.z);
        unsafeAtomicAdd(drow + 3, wt * v.w);
        if (lane == 0) unsafeAtomicAdd(dp2 + s, 1.0f);
    }
}

// ---------------------------------------------------------------------------
// Fused node update (one side per launch):
//   acc = (-dp .* Y) @ M + T @ A          (V_WMMA_F32_16X16X4_F32 chain)
//   out = relu((1-a)*Y + a*l/(1+l*deg) * (acc + X))
//
// Block = 256 threads = 8 waves. Each wave owns TWO 16x128 output stripes
// (rows m0.. and m0+128..), so every LDS B-fetch feeds 2 WMMAs.
// B is staged in LDS with k-pairs interleaved: lds2[p*128+col] =
// {B[2p][col], B[2p+1][col]}  ->  one aligned ds_load_b64 per B tile.
// ---------------------------------------------------------------------------
__global__ __launch_bounds__(256) void node_update(const float* __restrict__ Y,
                                                   const float* __restrict__ X,
                                                   const float* __restrict__ deg,
                                                   const float* __restrict__ dp,
                                                   const float* __restrict__ M,
                                                   const float* __restrict__ A,
                                                   const float* __restrict__ T,
                                                   const float* __restrict__ alp,
                                                   const float* __restrict__ lam,
                                                   float* __restrict__ out,
                                                   int N) {
    __shared__ v2f lds2[64 * D];   // 64 k-pairs x 128 cols, 64 KB

    const int lane = threadIdx.x & 31;
    const int wave = threadIdx.x >> 5;
    const int m0 = blockIdx.x * ROWS_PER_BLOCK + wave * 16;   // stripe 0
    const int m1 = m0 + 128;                                  // stripe 1

    // A-operand rows (ISA: lanes 0-15 M=lane K={0,1}; lanes 16-31 M=lane-16 K={2,3})
    const int rowA0 = min(m0 + (lane & 15), N - 1);
    const int rowA1 = min(m1 + (lane & 15), N - 1);
    const int koff = (lane < 16) ? 0 : 2;
    const int pairOff = koff >> 1;       // 0 for low lanes, 1 for high lanes
    const int colLow = lane & 15;

    v8f c0[8], c1[8];
#pragma unroll
    for (int t = 0; t < 8; ++t) {
        c0[t] = (v8f){0, 0, 0, 0, 0, 0, 0, 0};
        c1[t] = (v8f){0, 0, 0, 0, 0, 0, 0, 0};
    }

    // ---- stage 1: B = M (interleaved fill), A-operand = (-dp) * Y ------
    for (int idx = threadIdx.x; idx < 64 * D; idx += 256) {
        const int p = idx >> 7;
        const int col = idx & 127;
        v2f v;
        v.x = M[(2 * p) * D + col];
        v.y = M[(2 * p + 1) * D + col];
        lds2[idx] = v;
    }
    __syncthreads();

    {
        const float s0 = -dp[rowA0];
        const float s1 = -dp[rowA1];
        const float* y0 = Y + (long)rowA0 * D;
        const float* y1 = Y + (long)rowA1 * D;
#pragma unroll 2
        for (int kk = 0; kk < 32; ++kk) {
            const int k = kk * 4;
            v2f a0 = *(const v2f*)(y0 + k + koff);
            v2f a1 = *(const v2f*)(y1 + k + koff);
            a0 = a0 * s0;
            a1 = a1 * s1;
            const v2f* bbase = lds2 + (2 * kk + pairOff) * D + colLow;
#pragma unroll
            for (int t = 0; t < 8; ++t) {
                const v2f b = bbase[t * 16];
                c0[t] = __builtin_amdgcn_wmma_f32_16x16x4_f32(
                    false, a0, false, b, (short)0, c0[t], false, false);
                c1[t] = __builtin_amdgcn_wmma_f32_16x16x4_f32(
                    false, a1, false, b, (short)0, c1[t], false, false);
            }
        }
    }
    __syncthreads();

    // ---- stage 2: B = A (interleaved fill), A-operand = T --------------
    for (int idx = threadIdx.x; idx < 64 * D; idx += 256) {
        const int p = idx >> 7;
        const int col = idx & 127;
        v2f v;
        v.x = A[(2 * p) * D + col];
        v.y = A[(2 * p + 1) * D + col];
        lds2[idx] = v;
    }
    __syncthreads();

    {
        const float* t0 = T + (long)rowA0 * D;
        const float* t1 = T + (long)rowA1 * D;
#pragma unroll 2
        for (int kk = 0; kk < 32; ++kk) {
            const int k = kk * 4;
            const v2f a0 = *(const v2f*)(t0 + k + koff);
            const v2f a1 = *(const v2f*)(t1 + k + koff);
            const v2f* bbase = lds2 + (2 * kk + pairOff) * D + colLow;
#pragma unroll
            for (int t = 0; t < 8; ++t) {
                const v2f b = bbase[t * 16];
                c0[t] = __builtin_amdgcn_wmma_f32_16x16x4_f32(
                    false, a0, false, b, (short)0, c0[t], false, false);
                c1[t] = __builtin_amdgcn_wmma_f32_16x16x4_f32(
                    false, a1, false, b, (short)0, c1[t], false, false);
            }
        }
    }

    // ---- epilogue: +X, alpha/lambda blend, relu, masked store ----------
    const float av = alp[0];
    const float lv = lam[0];
#pragma unroll
    for (int st = 0; st < 2; ++st) {
        const v8f* c = st ? c1 : c0;
        const int rBase = (st ? m1 : m0) + ((lane >= 16) ? 8 : 0);
#pragma unroll
        for (int r = 0; r < 8; ++r) {
            const int row = rBase + r;
            if (row < N) {
                const float coef = av * lv / (1.0f + lv * deg[row]);
                const float* xrow = X + (long)row * D;
                const float* yrow = Y + (long)row * D;
                float* orow = out + (long)row * D;
#pragma unroll
                for (int t = 0; t < 8; ++t) {
                    const int col = colLow + t * 16;
                    const float ret = c[t][r] + xrow[col];
                    const float o = (1.0f - av) * yrow[col] + coef * ret;
                    orow[col] = fmaxf(o, 0.0f);
                }
            }
        }
    }
}

// ---------------------------------------------------------------------------
// Launcher
// ---------------------------------------------------------------------------
extern "C" void kernel_launch(void* const* d_in, const int* in_sizes, int n_in,
                              void* d_out, int out_size, void* d_ws, size_t ws_size,
                              hipStream_t stream) {
    const float* Y_user = (const float*)d_in[0];
    const float* Y_item = (const float*)d_in[1];
    const float* X_user = (const float*)d_in[2];
    const float* X_item = (const float*)d_in[3];
    const float* H_ui   = (const float*)d_in[4];
    const float* H_iu   = (const float*)d_in[5];
    const float* w1     = (const float*)d_in[6];
    const float* w2     = (const float*)d_in[7];
    const float* deg_u  = (const float*)d_in[8];
    const float* deg_i  = (const float*)d_in[9];
    const float* alp    = (const float*)d_in[10];
    const float* lam    = (const float*)d_in[11];
    const int*   src1   = (const int*)d_in[12];
    const int*   dst1   = (const int*)d_in[13];
    const int*   src2   = (const int*)d_in[14];
    const int*   dst2   = (const int*)d_in[15];

    const int N = in_sizes[0] / D;
    const int E = in_sizes[12];

    float* ws = (float*)d_ws;
    float* Muu   = ws;
    float* Mii   = Muu + D * D;
    float* A1    = Mii + D * D;
    float* A2    = A1 + D * D;
    float* dp1   = A2 + D * D;
    float* dp2   = dp1 + N;
    float* Titem = dp2 + N;
    float* Tuser = Titem + (long)N * D;

    float* out_u = (float*)d_out;               // relu(ret_u): [N, D]
    float* out_i = out_u + (long)N * D;         // relu(ret_i): [N, D]

    // 1) zero the accumulators (dp1, dp2, Titem, Tuser are contiguous)
    const long nzero = 2L * N + 2L * N * D;
    zero_f<<<8192, 256, 0, stream>>>(dp1, nzero);

    // 2) small matrix prep
    prep_mats<<<(D * D) / 256, 256, 0, stream>>>(H_ui, H_iu, Muu, Mii, A1, A2);

    // 3) edge scatter: one wave per edge, 2E edges total
    {
        const long threads = 2L * E * 32L;
        const int blocks = (int)((threads + 255) / 256);
        edge_scatter<<<blocks, 256, 0, stream>>>(Y_user, Y_item, src1, dst1,
                                                 src2, dst2, w1, w2,
                                                 dp1, dp2, Titem, Tuser, E);
    }

    // 4) fused WMMA node update, one launch per side
    const int nblk = (N + ROWS_PER_BLOCK - 1) / ROWS_PER_BLOCK;
    node_update<<<nblk, 256, 0, stream>>>(Y_user, X_user, deg_u, dp1,
                                          Muu, A2, Tuser, alp, lam, out_u, N);
    node_update<<<nblk, 256, 0, stream>>>(Y_item, X_item, deg_i, dp2,
                                          Mii, A1, Titem, alp, lam, out_i, N);
}